// PDCMBOLDFitter_56384330662221
// MI455X (gfx1250) — compile-verified
//
#include <hip/hip_runtime.h>
#include <stdint.h>

// ---------------- simulation constants (match reference) ----------------
#define NSTEPS   43400
#define NOBS     217
#define BENS     256
#define LANES    32
#define CHUNK    25                    // 25 | 43400, 25 | 200, 25 | 2000
#define NCHUNK   (NSTEPS / CHUNK)      // 1736

// LDS double buffer: [2][CHUNK][32 lanes][2 floats] = 12.8 KB per block
__global__ __launch_bounds__(LANES, 1)
void pdcm_bold_kernel(const float* __restrict__ p_sigma,
                      const float* __restrict__ p_mu,
                      const float* __restrict__ p_lamb,
                      const float* __restrict__ p_phi,
                      const float* __restrict__ p_chi,
                      const float* __restrict__ p_varphi,
                      const float* __restrict__ p_tMTT,
                      const float* __restrict__ p_tau,
                      const float* __restrict__ p_av,
                      const float* __restrict__ p_bv,
                      const float* __restrict__ p_ae,
                      const float* __restrict__ p_be,
                      const float* __restrict__ noise,   // [NSTEPS, 256, 2]
                      const float* __restrict__ obsn,    // [256, 217]
                      float* __restrict__ out)           // [256, 217]
{
    __shared__ float nbuf[2][CHUNK][LANES][2];

    const int lane   = threadIdx.x;
    const int member = blockIdx.x * LANES + lane;

    // uniform scalar parameters (single-element device arrays)
    const float sigma  = *p_sigma;
    const float mu     = *p_mu;
    const float lamb   = *p_lamb;
    const float phi    = *p_phi;
    const float chi    = *p_chi;
    const float varphi = *p_varphi;
    const float tMTT   = *p_tMTT;
    const float tau    = *p_tau;
    const float av     = *p_av;
    const float bv     = *p_bv;
    const float ae     = *p_ae;
    const float be     = *p_be;

    const float Hs          = 0.01f;
    const float w_s         = sqrtf(Hs) / sqrtf(fabsf(av * bv) + 1e-6f);
    const float w_o         = 0.01f / sqrtf(fabsf(ae * be) + 1e-6f);
    const float inv_alpha   = 3.125f;                       // f32(1/0.32)
    const float inv_mtt_tau = 1.0f / (tMTT + tau);
    const float inv_tmtt    = 1.0f / tMTT;
    const float log2_om_e0  = (float)(-0.7369655941662062); // log2(1 - E0)
    const float inv_E0      = 2.5f;                          // 1/0.4
    const float V0c = 0.04f, K1c = 2.8f, K2c = 2.0f, K3c = 0.6f;

    // LDS byte offsets of the two buffers for this lane (low 32 bits of the
    // flat shared address == wave-relative LDS address on gfx1250)
    const uint32_t lds0 = (uint32_t)(uintptr_t)(&nbuf[0][0][lane][0]);
    const uint32_t lds1 = (uint32_t)(uintptr_t)(&nbuf[1][0][lane][0]);
    const float* gbase  = noise + (size_t)member * 2;        // this lane's column

    // one async b64 per step-row: 32 lanes x 8B = 256 contiguous bytes
#define ISSUE_CHUNK(cidx, ldsbase)                                           \
    {                                                                        \
        const float* gp_ = gbase + (size_t)(cidx) * (CHUNK * BENS * 2);      \
        _Pragma("unroll")                                                    \
        for (int i_ = 0; i_ < CHUNK; ++i_) {                                 \
            asm volatile("global_load_async_to_lds_b64 %0, %1, off"          \
                         :: "v"((ldsbase) + (uint32_t)(i_ * LANES * 2 * 4)), \
                            "v"(gp_ + (size_t)i_ * (BENS * 2))               \
                         : "memory");                                        \
        }                                                                    \
    }

    ISSUE_CHUNK(0, lds0);

    float xe = 0.0f, xi = 0.0f, a = 0.0f, f = 1.0f, v = 1.0f, q = 1.0f;
    int obs_idx = 0;

    for (int c = 0; c < NCHUNK; ++c) {
        const int buf = c & 1;

        if (c + 1 < NCHUNK) {
            const uint32_t dst = (buf == 0) ? lds1 : lds0;
            ISSUE_CHUNK(c + 1, dst);
            // async loads complete in order: <=25 outstanding means the
            // just-issued chunk may still be in flight but chunk c is done
            asm volatile("s_wait_asynccnt 25" ::: "memory");
        } else {
            asm volatile("s_wait_asynccnt 0" ::: "memory");
        }

        // 20s ON / 40s OFF stimulus: u constant over each 25-step chunk
        const float u_t   = ((c % 240) < 80) ? 1.0f : 0.0f;
        const bool do_obs = ((c & 7) == 0);   // step c*25 % 200 == 0

#pragma unroll
        for (int i = 0; i < CHUNK; ++i) {
            const float n0 = nbuf[buf][i][lane][0];
            const float n1 = nbuf[buf][i][lane][1];

            // neuronal states (Euler-Maruyama), all RHS use OLD values
            const float xe_n = xe + Hs * (-sigma * xe - mu * xi + u_t) + w_s * n0;
            const float xi_n = xi + Hs * (lamb * (xe - xi)) + w_s * n1;
            // neurovascular coupling / inflow
            const float a_n = a + Hs * (xe - chi * a);
            const float f_n = f + Hs * (phi * a - varphi * (f - 1.0f));
            // balloon model
            const float fv   = __builtin_amdgcn_exp2f(inv_alpha * __builtin_amdgcn_logf(v));
            const float dv   = (f - fv) * inv_mtt_tau;
            const float v_n  = fmaxf(v + Hs * dv, 1e-4f);
            const float fout = fv + tau * dv;
            const float fsafe = fmaxf(f, 1e-4f);
            const float E    = 1.0f - __builtin_amdgcn_exp2f(log2_om_e0 * __builtin_amdgcn_rcpf(fsafe));
            const float dq   = (fsafe * E * inv_E0
                                - fout * q * __builtin_amdgcn_rcpf(v)) * inv_tmtt;
            const float q_n  = fmaxf(q + Hs * dq, 1e-4f);

            xe = xe_n; xi = xi_n; a = a_n; f = f_n; v = v_n; q = q_n;

            if (i == 0 && do_obs) {
                const float bold = V0c * (K1c * (1.0f - q)
                                   + K2c * (1.0f - q * __builtin_amdgcn_rcpf(v))
                                   + K3c * (1.0f - v));
                const int o = member * NOBS + obs_idx;
                out[o] = bold + w_o * obsn[o];
            }
        }
        if (do_obs) obs_idx++;
    }
#undef ISSUE_CHUNK
}

extern "C" void kernel_launch(void* const* d_in, const int* in_sizes, int n_in,
                              void* d_out, int out_size, void* d_ws, size_t ws_size,
                              hipStream_t stream) {
    (void)in_sizes; (void)n_in; (void)out_size; (void)d_ws; (void)ws_size;
    pdcm_bold_kernel<<<BENS / LANES, LANES, 0, stream>>>(
        (const float*)d_in[0],  (const float*)d_in[1],
        (const float*)d_in[2],  (const float*)d_in[3],
        (const float*)d_in[4],  (const float*)d_in[5],
        (const float*)d_in[6],  (const float*)d_in[7],
        (const float*)d_in[8],  (const float*)d_in[9],
        (const float*)d_in[10], (const float*)d_in[11],
        (const float*)d_in[12], (const float*)d_in[13],
        (float*)d_out);
}